// FGWF_25220047962323
// MI455X (gfx1250) — compile-verified
//
#include <hip/hip_runtime.h>
#include <stdint.h>

// ---------------------------------------------------------------------------
// Fused Gromov-Wasserstein inner product  <cost_mat, T>  for MI455X (gfx1250)
//   N = NB = 2048, D = 128.
//   - f32 WMMA (V_WMMA_F32_16X16X4_F32) for the two 2048^3 GEMM passes
//   - GLOBAL_LOAD_ASYNC_TO_LDS_B128/B32 double-buffered global->LDS staging
//     (ASYNCcnt pipelined one K-chunk ahead of the WMMA compute)
// ---------------------------------------------------------------------------

typedef __attribute__((ext_vector_type(2))) float v2f;
typedef __attribute__((ext_vector_type(8))) float v8f;

#define Nn 2048
#define Dd 128
#define LSTR 36   // padded LDS row stride (floats): 144B -> 16B aligned rows,
                  // (36*m) mod 64 banks distinct for m=0..15  -> conflict-free frags

__device__ __forceinline__ float wave_reduce(float s) {
#pragma unroll
    for (int off = 16; off > 0; off >>= 1)
        s += __shfl_xor(s, off, 32);
    return s;
}

// ---- CDNA5 async global->LDS (ASYNCcnt) -----------------------------------
__device__ __forceinline__ void async_b128(const void* lds, const float* g) {
    asm volatile("global_load_async_to_lds_b128 %0, %1, off"
                 :: "v"((uint32_t)(uintptr_t)lds), "v"(g) : "memory");
}
__device__ __forceinline__ void async_b32(const void* lds, const float* g) {
    asm volatile("global_load_async_to_lds_b32 %0, %1, off"
                 :: "v"((uint32_t)(uintptr_t)lds), "v"(g) : "memory");
}
template <int N> __device__ __forceinline__ void wait_async() {
    asm volatile("s_wait_asynccnt %0" :: "i"(N) : "memory");
}

// ---------------- small reductions ----------------

__global__ __launch_bounds__(256) void k_rowsum(const float* __restrict__ T,
                                                float* __restrict__ r) {
    const int lane = threadIdx.x & 31;
    const int row  = blockIdx.x * 8 + (threadIdx.x >> 5);
    float s = 0.f;
    for (int j = lane; j < Nn; j += 32) s += T[row * Nn + j];
    s = wave_reduce(s);
    if (lane == 0) r[row] = s;
}

__global__ __launch_bounds__(256) void k_colsum(const float* __restrict__ T,
                                                float* __restrict__ c) {
    const int j = blockIdx.x * 256 + threadIdx.x;
    float s = 0.f;
    for (int i = 0; i < Nn; ++i) s += T[i * Nn + j];
    c[j] = s;
}

// per row i:  ( sum_k C[i,k]^2 * p[k]  +  0.5*||E[i]||^2 ) * w[i]
__global__ __launch_bounds__(256) void k_diag(const float* __restrict__ C,
                                              const float* __restrict__ p,
                                              const float* __restrict__ E,
                                              const float* __restrict__ w,
                                              float* __restrict__ partial) {
    __shared__ float wpart[8];
    const int lane = threadIdx.x & 31;
    const int wave = threadIdx.x >> 5;
    const int row  = blockIdx.x * 8 + wave;
    float s = 0.f;
    for (int k = lane; k < Nn; k += 32) {
        float v = C[row * Nn + k];
        s += v * v * p[k];
    }
    float ns = 0.f;
    for (int d = lane; d < Dd; d += 32) {
        float e = E[row * Dd + d];
        ns += e * e;
    }
    float t = wave_reduce(s + 0.5f * ns);
    if (lane == 0) wpart[wave] = t * w[row];
    __syncthreads();
    if (threadIdx.x == 0) {
        float tt = 0.f;
#pragma unroll
        for (int i = 0; i < 8; ++i) tt += wpart[i];
        partial[blockIdx.x] = tt;
    }
}

// ---------------- WMMA inner step: one 32-deep K chunk ----------------------
// Wave owns 32x32 = 2x2 tiles of 16x16.  A stored [m][k], B stored [n][k] in
// LDS (both padded stride LSTR), so both fragments are plain 8-byte loads:
//   A frag (16x4): lanes 0-15 row M=lane, V0/V1 = K0/K1; lanes 16-31 K2/K3
//   B frag (4x16): lanes 0-15 col N=lane, V0/V1 = K0/K1; lanes 16-31 K2/K3
__device__ __forceinline__ void mma_chunk(const float (*As)[LSTR],
                                          const float (*Bs)[LSTR],
                                          int wm, int wn, int l15, int koff,
                                          v8f acc[2][2]) {
#pragma unroll
    for (int kk = 0; kk < 32; kk += 4) {
        v2f a0 = *(const v2f*)&As[wm * 32 + l15][kk + koff];
        v2f a1 = *(const v2f*)&As[wm * 32 + 16 + l15][kk + koff];
        v2f b0 = *(const v2f*)&Bs[wn * 32 + l15][kk + koff];
        v2f b1 = *(const v2f*)&Bs[wn * 32 + 16 + l15][kk + koff];
        acc[0][0] = __builtin_amdgcn_wmma_f32_16x16x4_f32(false, a0, false, b0,
                                                          (short)0, acc[0][0], false, false);
        acc[0][1] = __builtin_amdgcn_wmma_f32_16x16x4_f32(false, a0, false, b1,
                                                          (short)0, acc[0][1], false, false);
        acc[1][0] = __builtin_amdgcn_wmma_f32_16x16x4_f32(false, a1, false, b0,
                                                          (short)0, acc[1][0], false, false);
        acc[1][1] = __builtin_amdgcn_wmma_f32_16x16x4_f32(false, a1, false, b1,
                                                          (short)0, acc[1][1], false, false);
    }
}

// ---------------- pass 1:  C = alpha * A @ B  (2048x2048, row-major) --------
// Async pipeline: issue chunk c+1's global->LDS DMA, wait ASYNCcnt<=P (chunk c
// landed), barrier, WMMA on buffer c&1, barrier (protects buffer reuse).
__global__ __launch_bounds__(256) void k_gemm1(const float* __restrict__ A,
                                               const float* __restrict__ B,
                                               float* __restrict__ C,
                                               float alpha) {
    __shared__ float As[2][128][LSTR];
    __shared__ float Bs[2][64][LSTR];
    const int tid  = threadIdx.x;
    const int lane = tid & 31;
    const int wave = tid >> 5;
    const int wm   = wave & 3;
    const int wn   = wave >> 2;
    const int bm   = blockIdx.y * 128;
    const int bn   = blockIdx.x * 64;
    const int l15  = lane & 15;
    const int koff = (lane >> 4) << 1;

    // P = 4 (A b128) + 8 (B b32) = 12 async ops per thread per chunk
    auto issue = [&](int c, int buf) {
        const int kb = c * 32;
#pragma unroll
        for (int j = 0; j < 4; ++j) {               // A[m][k] <- A[bm+m][kb+k]
            int task = tid + 256 * j;
            int m = task >> 3, q = (task & 7) << 2;
            async_b128(&As[buf][m][q], A + (size_t)(bm + m) * Nn + kb + q);
        }
#pragma unroll
        for (int j = 0; j < 8; ++j) {               // Bs[n][k] <- B[kb+k][bn+n]
            int task = tid + 256 * j;
            int n = task & 63, k = task >> 6;       // global-coalesced in n
            async_b32(&Bs[buf][n][k], B + (size_t)(kb + k) * Nn + bn + n);
        }
    };

    v8f acc[2][2] = {};
    issue(0, 0);
    for (int c = 0; c < 64; ++c) {
        const int cur = c & 1;
        if (c + 1 < 64) { issue(c + 1, cur ^ 1); wait_async<12>(); }
        else            { wait_async<0>(); }
        __syncthreads();
        mma_chunk(As[cur], Bs[cur], wm, wn, l15, koff, acc);
        __syncthreads();
    }

    const int rbase = bm + wm * 32 + ((lane >> 4) << 3);
    const int cbase = bn + wn * 32 + l15;
#pragma unroll
    for (int ti = 0; ti < 2; ++ti)
#pragma unroll
        for (int tj = 0; tj < 2; ++tj)
#pragma unroll
            for (int v = 0; v < 8; ++v)
                C[(size_t)(rbase + ti * 16 + v) * Nn + cbase + tj * 16] = alpha * acc[ti][tj][v];
}

// ---------------- pass 2:  partial = -sum( (M2@Ct^T + Es@Et^T) .* T ) -------
// Both B sources are accessed transposed (B[k][n] = src[n][k]) -> the [n][k]
// LDS layout makes every async transfer a contiguous b128.
__global__ __launch_bounds__(256) void k_pass2(const float* __restrict__ M2,
                                               const float* __restrict__ Ct,
                                               const float* __restrict__ Es,
                                               const float* __restrict__ Et,
                                               const float* __restrict__ T,
                                               float* __restrict__ partial) {
    __shared__ float As[2][128][LSTR];
    __shared__ float Bs[2][64][LSTR];
    __shared__ float wpart[8];
    const int tid  = threadIdx.x;
    const int lane = tid & 31;
    const int wave = tid >> 5;
    const int wm   = wave & 3;
    const int wn   = wave >> 2;
    const int bm   = blockIdx.y * 128;
    const int bn   = blockIdx.x * 64;
    const int l15  = lane & 15;
    const int koff = (lane >> 4) << 1;

    const int NC = 64 + 4;   // 64 chunks of M2@Ct^T (K=2048) + 4 of Es@Et^T (K=128)

    // P = 4 (A b128) + 2 (B b128) = 6 async ops per thread per chunk
    auto issue = [&](int c, int buf) {
        const float* Ap; const float* Bp; int lda, ldb, kb;
        if (c < 64) { Ap = M2; lda = Nn; Bp = Ct; ldb = Nn; kb = c * 32; }
        else        { Ap = Es; lda = Dd; Bp = Et; ldb = Dd; kb = (c - 64) * 32; }
#pragma unroll
        for (int j = 0; j < 4; ++j) {               // As[m][k] <- Ap[bm+m][kb+k]
            int task = tid + 256 * j;
            int m = task >> 3, q = (task & 7) << 2;
            async_b128(&As[buf][m][q], Ap + (size_t)(bm + m) * lda + kb + q);
        }
#pragma unroll
        for (int j = 0; j < 2; ++j) {               // Bs[n][k] <- Bp[bn+n][kb+k]
            int task = tid + 256 * j;
            int n = task >> 3, q = (task & 7) << 2;
            async_b128(&Bs[buf][n][q], Bp + (size_t)(bn + n) * ldb + kb + q);
        }
    };

    v8f acc[2][2] = {};
    issue(0, 0);
    for (int c = 0; c < NC; ++c) {
        const int cur = c & 1;
        if (c + 1 < NC) { issue(c + 1, cur ^ 1); wait_async<6>(); }
        else            { wait_async<0>(); }
        __syncthreads();
        mma_chunk(As[cur], Bs[cur], wm, wn, l15, koff, acc);
        __syncthreads();
    }

    // fold against matching T tile and reduce
    const int rbase = bm + wm * 32 + ((lane >> 4) << 3);
    const int cbase = bn + wn * 32 + l15;
    float s = 0.f;
#pragma unroll
    for (int ti = 0; ti < 2; ++ti)
#pragma unroll
        for (int tj = 0; tj < 2; ++tj)
#pragma unroll
            for (int v = 0; v < 8; ++v)
                s += acc[ti][tj][v] * T[(size_t)(rbase + ti * 16 + v) * Nn + cbase + tj * 16];
    s = wave_reduce(s);
    if (lane == 0) wpart[wave] = s;
    __syncthreads();
    if (tid == 0) {
        float t = 0.f;
#pragma unroll
        for (int w = 0; w < 8; ++w) t += wpart[w];
        partial[blockIdx.y * gridDim.x + blockIdx.x] = -t;
    }
}

// ---------------- finalize (deterministic fixed-order sum) ------------------
__global__ void k_final(const float* __restrict__ p1, const float* __restrict__ p2,
                        const float* __restrict__ p3, float* __restrict__ out) {
    double a = 0.0;
    for (int i = 0; i < 256; ++i) a += (double)p1[i];
    for (int i = 0; i < 256; ++i) a += (double)p2[i];
    for (int i = 0; i < 512; ++i) a += (double)p3[i];
    out[0] = (float)a;
}

// ---------------------------------------------------------------------------

extern "C" void kernel_launch(void* const* d_in, const int* in_sizes, int n_in,
                              void* d_out, int out_size, void* d_ws, size_t ws_size,
                              hipStream_t stream) {
    (void)in_sizes; (void)n_in; (void)out_size; (void)ws_size;
    const float* Cs = (const float*)d_in[0];   // graph        (N,N)
    const float* Es = (const float*)d_in[1];   // embedding    (N,D)
    const float* ps = (const float*)d_in[2];   // prob         (N,1)
    const float* Ct = (const float*)d_in[3];   // graph_b      (NB,NB)
    const float* Et = (const float*)d_in[4];   // embedding_b  (NB,D)
    const float* pt = (const float*)d_in[5];   // prob_b       (NB,1)
    const float* T  = (const float*)d_in[6];   // tran         (N,NB)
    // d_in[7..9]: weights / ole_coeff / idx — do not enter the returned scalar.

    float* ws = (float*)d_ws;
    float* M  = ws;                        // 2048*2048 floats: 2*Cs@T
    float* r  = M + (size_t)Nn * Nn;       // row sums of T
    float* c  = r + Nn;                    // col sums of T
    float* p1 = c + Nn;                    // 256 block partials (f1 + emb_s)
    float* p2 = p1 + 256;                  // 256 block partials (f2 + emb_t)
    float* p3 = p2 + 256;                  // 512 block partials (cross terms)
    float* out = (float*)d_out;

    k_rowsum<<<256, 256, 0, stream>>>(T, r);
    k_colsum<<<8, 256, 0, stream>>>(T, c);
    k_diag<<<256, 256, 0, stream>>>(Cs, ps, Es, r, p1);
    k_diag<<<256, 256, 0, stream>>>(Ct, pt, Et, c, p2);

    dim3 grid(32, 16);                     // 64-wide N blocks x 128-tall M blocks
    k_gemm1<<<grid, 256, 0, stream>>>(Cs, T, M, 2.0f);
    k_pass2<<<grid, 256, 0, stream>>>(M, Ct, Es, Et, T, p3);

    k_final<<<1, 1, 0, stream>>>(p1, p2, p3, out);
}